// SpatialAffineLayer_26396869001334
// MI455X (gfx1250) — compile-verified
//
#include <hip/hip_runtime.h>
#include <hip/hip_bf16.h>

typedef __attribute__((ext_vector_type(4))) float f4;
typedef __attribute__((ext_vector_type(2))) float v2f;
typedef __attribute__((ext_vector_type(8))) float v8f;

// X layout: [B=32][H=256][W=256][C=16] float32, NHWC.
// ws layout (floats): [0 .. 32768)   partial sums [32][64][16]
//                     [32768 .. )    theta        [32][6]

#define B_N 32
#define HW 65536          // 256*256 pixels per image
#define CH 16
#define BLOCKS_PER_B 64   // reduction blocks per batch
#define PIX_PER_RBLK 1024 // 65536 / 64

// ---------------------------------------------------------------------------
// Kernel 1: per-block partial channel sums over H*W (deterministic, no atomics)
// ---------------------------------------------------------------------------
__global__ void stn_reduce_kernel(const float* __restrict__ X,
                                  float* __restrict__ partial) {
    const int t = threadIdx.x;           // 0..255
    const int b = blockIdx.y;            // batch
    const int blk = blockIdx.x;          // 0..63

    const f4* src = (const f4*)(X + (size_t)b * (HW * CH));
    const int base = blk * (PIX_PER_RBLK * CH / 4);  // f4 units: 4096 per block

    // Each thread strides by 256 f4's; channel group = t % 4 (constant since
    // stride 256 and base 4096 are multiples of 4).
    f4 acc = {0.f, 0.f, 0.f, 0.f};
#pragma unroll
    for (int i = 0; i < 16; ++i) {
        acc += src[base + i * 256 + t];
    }

    __shared__ float lds[256 * 4];
    ((f4*)lds)[t] = acc;
    __syncthreads();

    if (t < CH) {
        const int cg = t >> 2;   // which channel-group (j % 4 must equal this)
        const int k  = t & 3;    // element within float4
        float s = 0.f;
        for (int j = cg; j < 256; j += 4) s += lds[j * 4 + k];
        partial[((size_t)b * BLOCKS_PER_B + blk) * CH + t] = s;
    }
}

// ---------------------------------------------------------------------------
// Kernel 2: finish mean, theta = mean @ W_loc + b_loc   (WMMA f32 16x16x4)
//   A = mean [32 x 16]  (two 16-row M-tiles, four K=4 tiles)
//   B = W_loc [16 x 6] padded to 16 x 16 with zeros
// ---------------------------------------------------------------------------
__global__ void stn_theta_kernel(const float* __restrict__ partial,
                                 const float* __restrict__ W_loc,
                                 const float* __restrict__ b_loc,
                                 float* __restrict__ theta) {
    __shared__ float P[B_N * CH];        // pooled sums [32][16]
    const int t = threadIdx.x;           // 0..511
    {
        const int b = t >> 4;            // 0..31
        const int c = t & 15;            // 0..15
        float s = 0.f;
        for (int j = 0; j < BLOCKS_PER_B; ++j)
            s += partial[((size_t)b * BLOCKS_PER_B + j) * CH + c];
        P[b * CH + c] = s;
    }
    __syncthreads();

    if (t < 32) {                        // wave 0 only: EXEC all-ones, wave32
        const int lane = t;
        const int n  = lane & 15;        // N column (only n<6 valid)
        const int hi = lane >> 4;        // half-wave selects K pair
        const float scale = 1.0f / 65536.0f;   // mean over H*W

        // B fragments: 4x16 tile per K-step.
        // VGPR0: lanes 0-15 -> K=k0, lanes 16-31 -> K=k0+2 ; VGPR1: K=k0+1 / k0+3
        v2f bf[4];
#pragma unroll
        for (int kt = 0; kt < 4; ++kt) {
            const int k0 = kt * 4 + hi * 2;
            v2f bb;
            bb.x = (n < 6) ? W_loc[(k0 + 0) * 6 + n] : 0.0f;
            bb.y = (n < 6) ? W_loc[(k0 + 1) * 6 + n] : 0.0f;
            bf[kt] = bb;
        }

#pragma unroll
        for (int m = 0; m < 2; ++m) {    // M-tiles: batches 0-15, 16-31
            v8f acc = {};
            const int row = m * 16 + n;  // A rows: M = lane%16 (both halves)
#pragma unroll
            for (int kt = 0; kt < 4; ++kt) {
                const int k0 = kt * 4 + hi * 2;
                v2f af;
                af.x = P[row * CH + k0 + 0] * scale;
                af.y = P[row * CH + k0 + 1] * scale;
                acc = __builtin_amdgcn_wmma_f32_16x16x4_f32(
                    /*neg_a=*/false, af, /*neg_b=*/false, bf[kt],
                    /*c_mod=*/(short)0, acc, /*reuse_a=*/false, /*reuse_b=*/false);
            }
            // D layout: VGPR v, lanes 0-15 -> M=v, lanes 16-31 -> M=v+8; N=lane%16
            if (n < 6) {
                const float bias = b_loc[n];
#pragma unroll
                for (int v = 0; v < 8; ++v) {
                    const int batch = m * 16 + v + hi * 8;
                    theta[batch * 6 + n] = acc[v] + bias;
                }
            }
        }
    }
}

// ---------------------------------------------------------------------------
// Kernel 3: affine grid + bilinear sample. 4 lanes per output pixel
// (float4 of channels each). Gathers hit L2 (X = 134MB < 192MB L2);
// output written with non-temporal stores to avoid evicting X.
// ---------------------------------------------------------------------------
__global__ void stn_sample_kernel(const float* __restrict__ X,
                                  const float* __restrict__ theta,
                                  float* __restrict__ out) {
    const int t = threadIdx.x;               // 0..255
    const int b = blockIdx.y;                // batch
    const int p = blockIdx.x * 64 + (t >> 2);// output pixel 0..65535
    const int q = t & 3;                     // channel quad 0..3

    const int ox = p & 255;
    const int oy = p >> 8;

    const float* th = theta + b * 6;
    const float t0 = th[0], t1 = th[1], t2 = th[2];
    const float t3 = th[3], t4 = th[4], t5 = th[5];

    const float gx = fmaf((float)ox, 2.0f / 255.0f, -1.0f);
    const float gy = fmaf((float)oy, 2.0f / 255.0f, -1.0f);

    const float sx = fmaf(t0, gx, fmaf(t1, gy, t2));
    const float sy = fmaf(t3, gx, fmaf(t4, gy, t5));

    const float x = (sx + 1.0f) * 0.5f * 255.0f;
    const float y = (sy + 1.0f) * 0.5f * 255.0f;

    const float x0f = fminf(fmaxf(floorf(x), 0.0f), 255.0f);
    const float x1f = fminf(x0f + 1.0f, 255.0f);
    const float y0f = fminf(fmaxf(floorf(y), 0.0f), 255.0f);
    const float y1f = fminf(y0f + 1.0f, 255.0f);

    const int x0 = (int)x0f, x1 = (int)x1f;
    const int y0 = (int)y0f, y1 = (int)y1f;

    const float wa = (x1f - x) * (y1f - y);
    const float wb = (x1f - x) * (y - y0f);
    const float wc = (x - x0f) * (y1f - y);
    const float wd = (x - x0f) * (y - y0f);

    const f4* Xb = (const f4*)(X + (size_t)b * (HW * CH));
    const f4 Ia = Xb[(y0 * 256 + x0) * 4 + q];
    const f4 Ib = Xb[(y1 * 256 + x0) * 4 + q];
    const f4 Ic = Xb[(y0 * 256 + x1) * 4 + q];
    const f4 Id = Xb[(y1 * 256 + x1) * 4 + q];

    f4 o;
    o.x = wa * Ia.x + wb * Ib.x + wc * Ic.x + wd * Id.x;
    o.y = wa * Ia.y + wb * Ib.y + wc * Ic.y + wd * Id.y;
    o.z = wa * Ia.z + wb * Ib.z + wc * Ic.z + wd * Id.z;
    o.w = wa * Ia.w + wb * Ib.w + wc * Ic.w + wd * Id.w;

    f4* dst = (f4*)out + ((size_t)b * HW + p) * 4 + q;
    __builtin_nontemporal_store(o, dst);
}

// ---------------------------------------------------------------------------
extern "C" void kernel_launch(void* const* d_in, const int* in_sizes, int n_in,
                              void* d_out, int out_size, void* d_ws, size_t ws_size,
                              hipStream_t stream) {
    (void)in_sizes; (void)n_in; (void)out_size; (void)ws_size;

    const float* X     = (const float*)d_in[0];   // [32,256,256,16]
    const float* W_loc = (const float*)d_in[1];   // [16,6]
    const float* b_loc = (const float*)d_in[2];   // [6]
    float* out = (float*)d_out;                   // [32,256,256,16]

    float* partial = (float*)d_ws;                // 32*64*16 floats (128 KB)
    float* theta   = partial + B_N * BLOCKS_PER_B * CH;  // 32*6 floats

    stn_reduce_kernel<<<dim3(BLOCKS_PER_B, B_N), 256, 0, stream>>>(X, partial);
    stn_theta_kernel<<<1, 512, 0, stream>>>(partial, W_loc, b_loc, theta);
    stn_sample_kernel<<<dim3(HW / 64, B_N), 256, 0, stream>>>(X, theta, out);
}